// UVCoarseAttention_87737591923353
// MI455X (gfx1250) — compile-verified
//
#include <hip/hip_runtime.h>
#include <math.h>

typedef __attribute__((ext_vector_type(2))) float v2f;
typedef __attribute__((ext_vector_type(8))) float v8f;

#define EMBED      256
#define NUM_DEPTH  128
#define IMG_H      64
#define IMG_W      64
#define NV         4096          /* IMG_H * IMG_W == num_query == num_value */
#define QTILE      16            /* queries per workgroup (one WMMA M-tile)  */
#define NTILES     (NV / 16)     /* 256 pixel tiles of 16                    */
#define WAVES      8             /* 256 threads = 8 wave32                   */
#define KSTEPS     (EMBED / 4)   /* 64 WMMA K-steps per 16x16 tile           */
#define SROW       4100          /* padded LDS row stride in floats          */
#define LDS_BYTES  (QTILE * SROW * 4)   /* 262400 B of the 320KB WGP budget */

__device__ __forceinline__ int iclamp(int v, int lo, int hi) {
    return v < lo ? lo : (v > hi ? hi : v);
}

// One workgroup per 16-query tile.
// Phase 1: S_tile[16][4096] = key_tile(16x256) @ value^T(256x4096) via
//          V_WMMA_F32_16X16X4_F32. A fragments (key tile) are hoisted into
//          128 VGPRs so the steady-state loop is 1 global_load_b64 + 1 WMMA.
// Phase 2: bilinear gather of S_tile scalars per (q,d), scale 1/sqrt(256),
//          add identity, store out[q*128+d].
extern "C" __global__ __launch_bounds__(256)
void uv_coarse_attn_fused(const float* __restrict__ key,      // [4096, 256]
                          const float* __restrict__ value,    // [4096, 256]
                          const float* __restrict__ identity, // [4096, 128]
                          const float* __restrict__ ref3d,    // [524288, 2]
                          float* __restrict__ out)            // [4096, 128]
{
    extern __shared__ float s_S[];                 // [QTILE][SROW]

    const int qbase = blockIdx.x * QTILE;
    const int lane  = threadIdx.x & 31;
    const int wave  = threadIdx.x >> 5;
    const int n     = lane & 15;                   // M row of A / N col of B,D
    const int khalf = lane >> 4;                   // K sub-pair selector

    // ---- Hoist the A operand (key tile) into registers: 64 x v2f -------
    // Lane layout of 16x4 f32 A: VGPR0/1 = A[m][2*khalf + {0,1}] at each k0.
    const float* __restrict__ aRow =
        key + (size_t)(qbase + n) * EMBED + 2 * khalf;
    v2f aFrag[KSTEPS];
#pragma unroll
    for (int k = 0; k < KSTEPS; ++k)
        aFrag[k] = *(const v2f*)(aRow + 4 * k);

    // Each wave owns 32 of the 256 pixel tiles of the S row-panel.
    for (int t = wave; t < NTILES; t += WAVES) {
        const int pbase = t * 16;
        const float* __restrict__ bRow =
            value + (size_t)(pbase + n) * EMBED + 2 * khalf;

        // Prefetch the next pixel tile this wave will touch (streams in L2).
        __builtin_prefetch(bRow + (size_t)WAVES * 16 * EMBED, 0, 1);

        v8f acc = {0.f, 0.f, 0.f, 0.f, 0.f, 0.f, 0.f, 0.f};
        // Fully unrolled: keeps aFrag[] indices constant (register-resident)
        // and lets the scheduler hoist/cluster the 64 B loads.
#pragma unroll
        for (int k = 0; k < KSTEPS; ++k) {
            v2f b = *(const v2f*)(bRow + 4 * k);    // B[4k+2*khalf +{0,1}][n]
            acc = __builtin_amdgcn_wmma_f32_16x16x4_f32(
                      /*neg_a=*/false, aFrag[k], /*neg_b=*/false, b,
                      /*c_mod=*/(short)0, acc,
                      /*reuse_a=*/false, /*reuse_b=*/false);
        }
        // D layout: VGPR r holds S[m = r + 8*khalf][pbase + n].
        // SROW=4100 => rows r and r+8 land 32 banks apart: conflict-free.
#pragma unroll
        for (int r = 0; r < 8; ++r)
            s_S[(size_t)(r + 8 * khalf) * SROW + pbase + n] = acc[r];
    }

    __syncthreads();

    // ---- Phase 2: bilinear gather epilogue (attention weights == 1.0) ----
    const float scale = 0.0625f;                   // 1/sqrt(EMBED)
    for (int i = threadIdx.x; i < QTILE * NUM_DEPTH; i += 256) {
        const int ql = i >> 7;                     // 0..15 local query
        const int d  = i & (NUM_DEPTH - 1);        // 0..127 depth
        const size_t l = (size_t)(qbase + ql) * NUM_DEPTH + d;

        // px = ref_x*W - 0.5 ; py = ref_y*H - 0.5  (grid_sample, align=False)
        const float px = ref3d[2 * l + 0] * (float)IMG_W - 0.5f;
        const float py = ref3d[2 * l + 1] * (float)IMG_H - 0.5f;
        const float x0f = floorf(px), y0f = floorf(py);
        const int ix0 = (int)x0f, iy0 = (int)y0f;
        const int ix1 = ix0 + 1,  iy1 = iy0 + 1;
        const float wx1 = px - x0f, wx0 = 1.0f - wx1;
        const float wy1 = py - y0f, wy0 = 1.0f - wy1;

        const float mx0 = (ix0 >= 0 && ix0 < IMG_W) ? 1.0f : 0.0f;
        const float mx1 = (ix1 >= 0 && ix1 < IMG_W) ? 1.0f : 0.0f;
        const float my0 = (iy0 >= 0 && iy0 < IMG_H) ? 1.0f : 0.0f;
        const float my1 = (iy1 >= 0 && iy1 < IMG_H) ? 1.0f : 0.0f;

        const int cx0 = iclamp(ix0, 0, IMG_W - 1);
        const int cx1 = iclamp(ix1, 0, IMG_W - 1);
        const int cy0 = iclamp(iy0, 0, IMG_H - 1);
        const int cy1 = iclamp(iy1, 0, IMG_H - 1);

        const float* __restrict__ row = s_S + (size_t)ql * SROW;
        float acc = 0.0f;
        acc = fmaf(row[cy0 * IMG_W + cx0] * (my0 * mx0), wy0 * wx0, acc);
        acc = fmaf(row[cy0 * IMG_W + cx1] * (my0 * mx1), wy0 * wx1, acc);
        acc = fmaf(row[cy1 * IMG_W + cx0] * (my1 * mx0), wy1 * wx0, acc);
        acc = fmaf(row[cy1 * IMG_W + cx1] * (my1 * mx1), wy1 * wx1, acc);

        out[l] = fmaf(acc, scale, identity[l]);
    }
}

extern "C" void kernel_launch(void* const* d_in, const int* in_sizes, int n_in,
                              void* d_out, int out_size, void* d_ws, size_t ws_size,
                              hipStream_t stream) {
    // setup_inputs order:
    // 0:query 1:key 2:value 3:identity 4:ref_3d 5:spatial_shapes 6:W_attn 7:b_attn
    // Softmax over a singleton axis => attention weights are exactly 1.0, so
    // query / W_attn / b_attn are dead; spatial_shapes is the fixed 64x64.
    const float* key      = (const float*)d_in[1];
    const float* value    = (const float*)d_in[2];
    const float* identity = (const float*)d_in[3];
    const float* ref3d    = (const float*)d_in[4];
    float* out            = (float*)d_out;

    (void)in_sizes; (void)n_in; (void)out_size; (void)d_ws; (void)ws_size;

    // 256.2KB dynamic LDS per workgroup (320KB available per WGP on CDNA5).
    hipFuncSetAttribute((const void*)uv_coarse_attn_fused,
                        hipFuncAttributeMaxDynamicSharedMemorySize, LDS_BYTES);

    dim3 grid(NV / QTILE);   // 256 workgroups
    dim3 block(256);         // 8 wave32
    uv_coarse_attn_fused<<<grid, block, LDS_BYTES, stream>>>(
        key, value, identity, ref3d, out);
}